// FeatureAlign_V2_58437325029455
// MI455X (gfx1250) — compile-verified
//
#include <hip/hip_runtime.h>

// Problem constants (from reference)
#define Bk   2
#define CSk  128
#define CDk  128
#define Dk   16
#define Hk   32
#define Wk   32
#define Pk   (Dk*Hk*Wk)      // 16384
#define SD   8
#define SH   16
#define SW   16
#define Gk   8
#define Cgk  16
#define K3k  27
#define KOFF  648            // 27*3*8 offset channels
#define KOFFP 672            // padded to 42 tiles of 16 (21 m-tile pairs)
#define KCAT  256            // concat channels
#define NT    64             // voxel tile (N)
#define LDN   65             // offT stride  (odd -> conflict-free b32)
#define LDC   260            // catT/sampT stride (even, %64==4 -> b64 ok)
#define LDA   132            // wA stride        (even, %64==4 -> b64 ok)

typedef float v2f __attribute__((ext_vector_type(2)));
typedef float v4f __attribute__((ext_vector_type(4)));
typedef float v8f __attribute__((ext_vector_type(8)));

__device__ __forceinline__ v8f wmma_f32(v2f a, v2f b, v8f c) {
  // D = A(16x4 f32) * B(4x16 f32) + C(16x16 f32)  -> v_wmma_f32_16x16x4_f32
  return __builtin_amdgcn_wmma_f32_16x16x4_f32(false, a, false, b, (short)0, c,
                                               false, false);
}

__device__ __forceinline__ int iclamp(int v, int lo, int hi) {
  return v < lo ? lo : (v > hi ? hi : v);
}

// ---------------------------------------------------------------------------
// Kernel 0: weight prep.
//  - wpad[672][256]: zero-padded w_offset; columns >=128 pre-scaled by 2
//    (folds the "2*feat_up" concat scale into the weight).
//  - wdcnT[27][128][128]: w_dcn transposed to k-major (coalesced per-tap load)
// ---------------------------------------------------------------------------
__global__ __launch_bounds__(256) void prep_kernel(
    const float* __restrict__ w_off, const float* __restrict__ w_dcn,
    float* __restrict__ wpad, float* __restrict__ wdcnT) {
  int idx = blockIdx.x * 256 + threadIdx.x;
  const int NPAD = KOFFP * KCAT;        // 172032
  const int NTW  = K3k * CDk * CSk;     // 442368
  if (idx < NPAD) {
    int o = idx >> 8;
    int c = idx & 255;
    float s = (c >= CDk) ? 2.0f : 1.0f;
    wpad[idx] = (o < KOFF) ? s * w_off[o * KCAT + c] : 0.0f;
  } else if (idx < NPAD + NTW) {
    int t = idx - NPAD;                 // t = k*16384 + (o*128+c)
    int k = t >> 14;
    int r = t & 16383;
    wdcnT[t] = w_dcn[(long)r * K3k + k];
  }
}

// ---------------------------------------------------------------------------
// Kernel 1: 2x trilinear upsample, half-pixel centers, edge-clamped
// (equivalent to jax.image.resize for 2x). [B,CS,8,16,16]->[B,CS,16,32,32]
// ---------------------------------------------------------------------------
__global__ __launch_bounds__(256) void upsample2x_kernel(
    const float* __restrict__ src, float* __restrict__ up) {
  int idx = blockIdx.x * 256 + threadIdx.x;
  if (idx >= Bk * CSk * Pk) return;
  int x  = idx & (Wk - 1);
  int y  = (idx >> 5) & (Hk - 1);
  int z  = (idx >> 10) & (Dk - 1);
  int bc = idx >> 14;                       // b*CS + c
  float zc = (float)z * 0.5f - 0.25f;
  float yc = (float)y * 0.5f - 0.25f;
  float xc = (float)x * 0.5f - 0.25f;
  float z0f = floorf(zc), y0f = floorf(yc), x0f = floorf(xc);
  float fz = zc - z0f, fy = yc - y0f, fx = xc - x0f;
  int z0 = iclamp((int)z0f, 0, SD - 1), z1 = iclamp((int)z0f + 1, 0, SD - 1);
  int y0 = iclamp((int)y0f, 0, SH - 1), y1 = iclamp((int)y0f + 1, 0, SH - 1);
  int x0 = iclamp((int)x0f, 0, SW - 1), x1 = iclamp((int)x0f + 1, 0, SW - 1);
  const float* s = src + (long)bc * (SD * SH * SW);
  float v000 = s[(z0 * SH + y0) * SW + x0];
  float v001 = s[(z0 * SH + y0) * SW + x1];
  float v010 = s[(z0 * SH + y1) * SW + x0];
  float v011 = s[(z0 * SH + y1) * SW + x1];
  float v100 = s[(z1 * SH + y0) * SW + x0];
  float v101 = s[(z1 * SH + y0) * SW + x1];
  float v110 = s[(z1 * SH + y1) * SW + x0];
  float v111 = s[(z1 * SH + y1) * SW + x1];
  float c00 = v000 + (v001 - v000) * fx;
  float c01 = v010 + (v011 - v010) * fx;
  float c10 = v100 + (v101 - v100) * fx;
  float c11 = v110 + (v111 - v110) * fx;
  float c0 = c00 + (c01 - c00) * fy;
  float c1 = c10 + (c11 - c10) * fy;
  up[idx] = c0 + (c1 - c0) * fz;
}

// ---------------------------------------------------------------------------
// Kernel 2: fused offset-GEMM + deformable sampling + DCN-GEMM + ReLU.
// One workgroup: batch b, 64 consecutive output voxels. 256 threads = 8 waves.
// LDS region order puts activation tiles at base 0 so most fragment loads
// encode their byte offset in the DS immediate field.
// ---------------------------------------------------------------------------
__global__ __launch_bounds__(256) void falign_fused_kernel(
    const float* __restrict__ up,     // [B,CS,D,H,W]
    const float* __restrict__ dst,    // [B,CD,D,H,W]
    const float* __restrict__ wpad,   // [672,256] padded+prescaled
    const float* __restrict__ wdcnT,  // [27,128,128]
    float* __restrict__ out) {        // [B,CD,D,H,W]
  extern __shared__ float lds[];
  float* catT  = lds;                      // [NT][KCAT] stride LDC (base 0)
  float* sampT = lds;                      // [NT][CSk]  stride LDC (reused)
  float* wA    = lds + NT * LDC;           // [CDk][CSk] stride LDA
  float* offT  = wA + CDk * LDA;           // [KOFFP][NT] stride LDN

  const int tid  = threadIdx.x;
  const int lane = tid & 31;
  const int wave = tid >> 5;
  const bool hi  = lane >= 16;
  const int ln   = lane & 15;
  const int b    = blockIdx.y;
  const int p0   = blockIdx.x * NT;

  // ---- Stage A: cat tile = [dst(128); up(128)], b128 loads, transposed ----
  for (int i = tid; i < KCAT * (NT / 4); i += 256) {   // 4096 float4s
    int c = i >> 4;
    int q = i & 15;
    v4f v;
    if (c < CDk) v = *(const v4f*)&dst[((long)(b * CDk + c)) * Pk + p0 + q * 4];
    else         v = *(const v4f*)&up[((long)(b * CSk + (c - CDk))) * Pk + p0 + q * 4];
    int n = q * 4;
    catT[(n + 0) * LDC + c] = v[0];
    catT[(n + 1) * LDC + c] = v[1];
    catT[(n + 2) * LDC + c] = v[2];
    catT[(n + 3) * LDC + c] = v[3];
  }
  __syncthreads();

  // ---- Stage B: offset GEMM  offT[o,n] = sum_c wpad[o,c] * cat[c,n] -------
  // M = 672 (21 pairs of m-tiles), K = 256 (64 k-steps), N = 64 (4 n-tiles).
  // Two m-tiles per pass share the 4 B-fragments -> half the LDS B traffic.
  for (int mp = wave; mp < KOFFP / 32; mp += 8) {
    v8f a0 = {}, a1 = {}, a2 = {}, a3 = {};
    v8f a4 = {}, a5 = {}, a6 = {}, a7 = {};
    const float* arow0 = wpad + (long)(mp * 32 + ln) * KCAT;
    const float* arow1 = arow0 + 16 * KCAT;
#pragma unroll 2
    for (int kb = 0; kb < KCAT / 4; ++kb) {
      int c0 = kb * 4 + (hi ? 2 : 0);
      v2f av0 = *(const v2f*)(arow0 + c0);               // global_load_b64
      v2f av1 = *(const v2f*)(arow1 + c0);
      v2f b0 = *(const v2f*)&catT[( 0 + ln) * LDC + c0]; // ds_load_b64 x4
      v2f b1 = *(const v2f*)&catT[(16 + ln) * LDC + c0];
      v2f b2 = *(const v2f*)&catT[(32 + ln) * LDC + c0];
      v2f b3 = *(const v2f*)&catT[(48 + ln) * LDC + c0];
      a0 = wmma_f32(av0, b0, a0);
      a1 = wmma_f32(av0, b1, a1);
      a2 = wmma_f32(av0, b2, a2);
      a3 = wmma_f32(av0, b3, a3);
      a4 = wmma_f32(av1, b0, a4);
      a5 = wmma_f32(av1, b1, a5);
      a6 = wmma_f32(av1, b2, a6);
      a7 = wmma_f32(av1, b3, a7);
    }
    int ob0 = mp * 32 + (hi ? 8 : 0);   // D layout: vgpr r -> row r (+8 hi)
    int ob1 = ob0 + 16;
#pragma unroll
    for (int r = 0; r < 8; ++r) {
      offT[(ob0 + r) * LDN +  0 + ln] = a0[r];
      offT[(ob0 + r) * LDN + 16 + ln] = a1[r];
      offT[(ob0 + r) * LDN + 32 + ln] = a2[r];
      offT[(ob0 + r) * LDN + 48 + ln] = a3[r];
      offT[(ob1 + r) * LDN +  0 + ln] = a4[r];
      offT[(ob1 + r) * LDN + 16 + ln] = a5[r];
      offT[(ob1 + r) * LDN + 32 + ln] = a6[r];
      offT[(ob1 + r) * LDN + 48 + ln] = a7[r];
    }
  }
  __syncthreads();

  // ---- Stage C: per-tap sampling + DCN GEMM accumulation ------------------
  v8f o0 = {}, o1 = {}, o2 = {}, o3 = {};
  const float* upb = up + (long)b * CSk * Pk;
  for (int k = 0; k < K3k; ++k) {
    // stage w_dcn[:,:,k] (k-major layout) into LDS: b128 in, b128 out
    const float* wk = wdcnT + (long)k * (CDk * CSk);
    if (k + 1 < K3k) __builtin_prefetch(wk + CDk * CSk, 0, 0);
    for (int i = tid; i < CDk * (CSk / 4); i += 256) {   // 4096 float4s
      int o = i >> 5;
      int q = i & 31;
      *(v4f*)&wA[o * LDA + q * 4] = *(const v4f*)&wk[o * CSk + q * 4];
    }
    // deformable trilinear sampling (zero-padded) into sampT[n][c]
    int kz = k / 9 - 1, ky = (k / 3) % 3 - 1, kx = k % 3 - 1;
#pragma unroll
    for (int half = 0; half < 2; ++half) {
      int i = tid + half * 256;                 // 512 (g,n) pairs
      int g = i >> 6, n = i & (NT - 1);
      int p = p0 + n;
      int px = p & 31, py = (p >> 5) & 31, pz = p >> 10;
      int obch = (g * K3k + k) * 3;
      float zc = (float)(pz + kz) + offT[(obch + 0) * LDN + n];
      float yc = (float)(py + ky) + offT[(obch + 1) * LDN + n];
      float xc = (float)(px + kx) + offT[(obch + 2) * LDN + n];
      float z0f = floorf(zc), y0f = floorf(yc), x0f = floorf(xc);
      float fz = zc - z0f, fy = yc - y0f, fx = xc - x0f;
      int z0 = (int)z0f, y0 = (int)y0f, x0 = (int)x0f;
      float wgt[8]; int ofs[8];
#pragma unroll
      for (int j = 0; j < 8; ++j) {
        int dz = (j >> 2) & 1, dy = (j >> 1) & 1, dx = j & 1;
        int iz = z0 + dz, iy = y0 + dy, ix = x0 + dx;
        bool v = (iz >= 0) && (iz < Dk) && (iy >= 0) && (iy < Hk) &&
                 (ix >= 0) && (ix < Wk);
        float wv = (dz ? fz : 1.0f - fz) * (dy ? fy : 1.0f - fy) *
                   (dx ? fx : 1.0f - fx);
        wgt[j] = v ? wv : 0.0f;
        ofs[j] = (iclamp(iz, 0, Dk - 1) * Hk + iclamp(iy, 0, Hk - 1)) * Wk +
                 iclamp(ix, 0, Wk - 1);
      }
      const float* chan = upb + (long)(g * Cgk) * Pk;
#pragma unroll
      for (int c4 = 0; c4 < 4; ++c4) {          // 4 channels -> ds_store_b128
        v4f sv;
#pragma unroll
        for (int u = 0; u < 4; ++u) {
          const float* cp = chan + (long)(c4 * 4 + u) * Pk;
          float acc = 0.0f;
#pragma unroll
          for (int j = 0; j < 8; ++j) acc = fmaf(wgt[j], cp[ofs[j]], acc);
          sv[u] = acc;
        }
        *(v4f*)&sampT[n * LDC + g * Cgk + c4 * 4] = sv;
      }
    }
    __syncthreads();
    // GEMM: out_tile += wA(128x128) * samp(128x64); wave -> 16-row slab
    int orow = wave * 16 + ln;
#pragma unroll 4
    for (int kb = 0; kb < CSk / 4; ++kb) {
      int c0 = kb * 4 + (hi ? 2 : 0);
      v2f av = *(const v2f*)&wA[orow * LDA + c0];          // ds_load_b64
      v2f b0 = *(const v2f*)&sampT[( 0 + ln) * LDC + c0];
      v2f b1 = *(const v2f*)&sampT[(16 + ln) * LDC + c0];
      v2f b2 = *(const v2f*)&sampT[(32 + ln) * LDC + c0];
      v2f b3 = *(const v2f*)&sampT[(48 + ln) * LDC + c0];
      o0 = wmma_f32(av, b0, o0);
      o1 = wmma_f32(av, b1, o1);
      o2 = wmma_f32(av, b2, o2);
      o3 = wmma_f32(av, b3, o3);
    }
    __syncthreads();  // protect sampT/wA before next tap overwrites
  }

  // ---- Epilogue: ReLU + store --------------------------------------------
  int obase = wave * 16 + (hi ? 8 : 0);
  float* op = out + (long)(b * CDk) * Pk + p0;
#pragma unroll
  for (int r = 0; r < 8; ++r) {
    op[(long)(obase + r) * Pk +  0 + ln] = fmaxf(o0[r], 0.0f);
    op[(long)(obase + r) * Pk + 16 + ln] = fmaxf(o1[r], 0.0f);
    op[(long)(obase + r) * Pk + 32 + ln] = fmaxf(o2[r], 0.0f);
    op[(long)(obase + r) * Pk + 48 + ln] = fmaxf(o3[r], 0.0f);
  }
}

extern "C" void kernel_launch(void* const* d_in, const int* in_sizes, int n_in,
                              void* d_out, int out_size, void* d_ws, size_t ws_size,
                              hipStream_t stream) {
  const float* feat_1x = (const float*)d_in[0];   // [2,128,8,16,16]
  const float* feat_2x = (const float*)d_in[1];   // [2,128,16,32,32]
  const float* w_off   = (const float*)d_in[2];   // [648,256]
  const float* w_dcn   = (const float*)d_in[3];   // [128,128,3,3,3]
  float* out = (float*)d_out;

  // workspace layout (floats): up | wpad | wdcnT   (~18.4 MB total)
  float* up    = (float*)d_ws;                       // 2*128*16384
  float* wpad  = up + (size_t)Bk * CSk * Pk;         // 672*256
  float* wdcnT = wpad + (size_t)KOFFP * KCAT;        // 27*128*128

  int prep_total = KOFFP * KCAT + K3k * CDk * CSk;   // 614400
  prep_kernel<<<(prep_total + 255) / 256, 256, 0, stream>>>(w_off, w_dcn,
                                                            wpad, wdcnT);
  int total = Bk * CSk * Pk;
  upsample2x_kernel<<<(total + 255) / 256, 256, 0, stream>>>(feat_1x, up);

  dim3 grid(Pk / NT, Bk);                            // (256, 2)
  size_t shbytes =
      (size_t)(NT * LDC + CDk * LDA + KOFFP * LDN) * sizeof(float); // ~302 KB
  falign_fused_kernel<<<grid, 256, shbytes, stream>>>(up, feat_2x, wpad,
                                                      wdcnT, out);
}